// ChamferLossLayer_6330781794837
// MI455X (gfx1250) — compile-verified
//
#include <hip/hip_runtime.h>

// CDNA5 / gfx1250 Chamfer distance via V_WMMA_F32_16X16X4_F32.
//
// d[i,j] = |a_i|^2 + |b_j|^2 - 2 a_i . b_j
// A[i,:] = (-2ax,-2ay,-2az, 1), B[:,j] = (bx,by,bz,|b_j|^2)
//   => (A.B)[i,j] = -2 a.b + |b_j|^2 ; add |a_i|^2 after the row-min.
//
// Fragment layouts (ISA 7.12.2, 32-bit operands, wave32):
//   A 16x4: lane l, VGPR v -> (M = l%16, K = 2*(l/16)+v)
//   B 4x16: lane l, VGPR v -> (K = 2*(l/16)+v, N = l%16)
//   C/D 16x16: lane l, VGPR r -> (M = r + 8*(l/16), N = l%16)
// So each lane loads one 8-byte pair of its point's packed (x,y,z,sq).

typedef __attribute__((ext_vector_type(2))) float v2f;
typedef __attribute__((ext_vector_type(8))) float v8f;

__global__ void chamfer_zero_kernel(float* __restrict__ out, int n) {
    int t = blockIdx.x * blockDim.x + threadIdx.x;
    if (t < n) out[t] = 0.0f;
}

// Gather sampled points and pack as (x, y, z, |p|^2) per sample.
__global__ void chamfer_gather_kernel(const float* __restrict__ cloud,
                                      const int* __restrict__ idx,
                                      float4* __restrict__ pk,
                                      int P, int S, int NS) {
    int t = blockIdx.x * blockDim.x + threadIdx.x;
    if (t >= NS) return;
    int n = t / S;
    int s = t - n * S;
    int i = idx[s];
    const float* p = cloud + ((size_t)n * P + (size_t)i) * 3;
    float x = p[0], y = p[1], z = p[2];
    pk[t] = make_float4(x, y, z, x * x + y * y + z * z);
}

// One wave per (batch, 16-row block, direction). blockDim = 256 (8 waves).
__global__ __launch_bounds__(256)
void chamfer_min_kernel(const float4* __restrict__ pk1,
                        const float4* __restrict__ pk2,
                        float* __restrict__ out, int S, int N) {
    const int lane = threadIdx.x & 31;
    const int wave = threadIdx.x >> 5;
    const int rowBlocksPerBatch = S >> 4;

    const int gbr = blockIdx.x * 8 + wave;           // global row-block id (wave-uniform)
    if (gbr >= N * rowBlocksPerBatch) return;        // whole wave exits together
    const int n  = gbr / rowBlocksPerBatch;
    const int i0 = (gbr - n * rowBlocksPerBatch) << 4;

    // Direction: y==0 -> A=s1,B=s2 (a_to_b); y==1 -> A=s2,B=s1 (b_to_a)
    const float4* Apk = (blockIdx.y == 0) ? pk1 : pk2;
    const float4* Bpk = (blockIdx.y == 0) ? pk2 : pk1;

    const int g  = lane >> 4;     // component-pair group: 0 -> (x,y), 1 -> (z,w)
    const int lp = lane & 15;     // row/col within tile

    // ---- A fragment: rows i0..i0+15 of this batch, held for all 256 tiles ----
    const float* abase = (const float*)(Apk + (size_t)n * S + i0 + lp);
    v2f ap = *(const v2f*)(abase + (g << 1));        // g=0:(x,y)  g=1:(z,sq)
    v2f a;
    a.x = -2.0f * ap.x;
    a.y = g ? 1.0f : (-2.0f * ap.y);                 // K=3 slot carries the affine 1

    float runmin[8];
#pragma unroll
    for (int r = 0; r < 8; ++r) runmin[r] = 3.0e38f;

    const float* bcol = (const float*)(Bpk + (size_t)n * S);
    const v8f cz = {};                               // C = 0 each tile (no carry)

    for (int j0 = 0; j0 < S; j0 += 16) {
        // ---- B fragment: columns j0..j0+15 ----
        v2f b = *(const v2f*)(bcol + (size_t)(j0 + lp) * 4 + (g << 1));

        // D = A x B : d[r] is (-2 a.b + |b|^2) for (M = r + 8g, N = lp)
        v8f d = __builtin_amdgcn_wmma_f32_16x16x4_f32(
            /*neg_a=*/false, a, /*neg_b=*/false, b,
            /*c_mod=*/(short)0, cz, /*reuse_a=*/false, /*reuse_b=*/false);

#pragma unroll
        for (int r = 0; r < 8; ++r) runmin[r] = fminf(runmin[r], d[r]);
    }

    // Row-min finalize: reduce across the 16 lanes sharing each M, add |a_i|^2.
    float sum = 0.0f;
#pragma unroll
    for (int r = 0; r < 8; ++r) {
        float v = runmin[r];
        v = fminf(v, __shfl_xor(v, 1, 32));
        v = fminf(v, __shfl_xor(v, 2, 32));
        v = fminf(v, __shfl_xor(v, 4, 32));
        v = fminf(v, __shfl_xor(v, 8, 32));
        if (lp == 0) {
            float sq = Apk[(size_t)n * S + i0 + r + 8 * g].w;   // |a_i|^2
            sum += v + sq;
        }
    }
    sum += __shfl_xor(sum, 16, 32);                  // combine M=0..7 and M=8..15 halves
    if (lane == 0) atomicAdd(&out[n], sum / (float)S);
}

extern "C" void kernel_launch(void* const* d_in, const int* in_sizes, int n_in,
                              void* d_out, int out_size, void* d_ws, size_t ws_size,
                              hipStream_t stream) {
    const float* cloud1 = (const float*)d_in[0];
    const float* cloud2 = (const float*)d_in[1];
    const int*   idx1   = (const int*)d_in[2];
    const int*   idx2   = (const int*)d_in[3];

    const int N = out_size;                 // 8
    const int S = in_sizes[2];              // 4096
    const int P = in_sizes[0] / (3 * N);    // 65536
    const int NS = N * S;

    float4* pk1 = (float4*)d_ws;            // N*S*16 B
    float4* pk2 = pk1 + (size_t)NS;         // N*S*16 B   (total 1 MB of ws)

    chamfer_zero_kernel<<<1, 64, 0, stream>>>((float*)d_out, N);

    chamfer_gather_kernel<<<(NS + 255) / 256, 256, 0, stream>>>(cloud1, idx1, pk1, P, S, NS);
    chamfer_gather_kernel<<<(NS + 255) / 256, 256, 0, stream>>>(cloud2, idx2, pk2, P, S, NS);

    const int rowBlocks = N * (S / 16);     // 2048 waves per direction
    dim3 grid((rowBlocks + 7) / 8, 2);      // 8 waves per 256-thread block; y = direction
    chamfer_min_kernel<<<grid, 256, 0, stream>>>(pk1, pk2, (float*)d_out, S, N);
}